// CombinedLoss_21028159881359
// MI455X (gfx1250) — compile-verified
//
#include <hip/hip_runtime.h>
#include <hip/hip_bf16.h>
#include <math.h>

typedef float v2f __attribute__((ext_vector_type(2)));
typedef float v8f __attribute__((ext_vector_type(8)));

#define EPS_F 1e-10f
#define NB 256
#define NTILE_C 16          // 256 centers / 16 rows per tile
#define HW (240 * 320)      // 76800 targets per batch
#define NTILE_T (HW / 16)   // 4800 target tiles per batch
#define BATCH 8
#define FINF_BITS 0x7F800000u

// Workspace layout (4-byte units):
//  [0]=cnt [1]=sum (p-t)^2*m [2]=sum d*m [3]=sum d^2*m   (float, atomicAdd)
//  [4..11] chamfer target->center partial sums per batch (float, atomicAdd)
//  [16 .. 16 + BATCH*NB) per-center min-over-targets, f32 bits (uint, atomicMin)
#define WS_CMIN 16

__global__ void init_ws_kernel(unsigned int* __restrict__ ws) {
    int i = threadIdx.x;
    if (i < 16) ws[i] = 0u;                    // 0.0f accumulators
    for (int j = i; j < BATCH * NB; j += blockDim.x)
        ws[WS_CMIN + j] = FINF_BITS;           // +inf
}

// ---------------------------------------------------------------------------
// Chamfer: pairwise squared-distance tiles via V_WMMA_F32_16X16X4_F32.
// D[i][j] = a_i^2*1 + a_i*(-2 b_j) + 1*b_j^2  (K=3, padded to 4).
// All min-accumulation is done on uint bit patterns (valid for d^2 >= 0),
// which lowers to v_min_u32 / v_min3_u32 with no float canonicalization.
// One wave per block; grid = (BATCH, waves-per-batch).
// ---------------------------------------------------------------------------
__global__ __launch_bounds__(32, 1)
void chamfer_kernel(const float* __restrict__ target,
                    const float* __restrict__ edges,
                    unsigned int* __restrict__ ws) {
    const int b    = blockIdx.x;
    const int lane = threadIdx.x & 31;
    const int l15  = lane & 15;
    const bool lo  = lane < 16;
    const int waveId        = blockIdx.y;
    const int wavesPerBatch = gridDim.y;

    // ---- A fragments: 16 center tiles, 2 VGPRs each -----------------------
    v2f afrag[NTILE_C];
    const float* eb = edges + b * (NB + 1);
    #pragma unroll
    for (int k = 0; k < NTILE_C; ++k) {
        int ci  = k * 16 + l15;
        float c = 0.5f * (eb[ci] + eb[ci + 1]);   // bin center
        // lanes 0-15: K=0 -> a^2, K=1 -> a ; lanes 16-31: K=2 -> 1, K=3 -> 0
        afrag[k].x = lo ? c * c : 1.0f;
        afrag[k].y = lo ? c     : 0.0f;
    }

    // running per-row (center) minima as uint bits, elementwise across tiles
    unsigned int rowAcc[NTILE_C * 8];
    #pragma unroll
    for (int i = 0; i < NTILE_C * 8; ++i) rowAcc[i] = FINF_BITS;

    const float* tb = target + (size_t)b * HW;
    float sumMin = 0.0f;
    const v8f czero = {0.f, 0.f, 0.f, 0.f, 0.f, 0.f, 0.f, 0.f};

    for (int t = waveId; t < NTILE_T; t += wavesPerBatch) {
        float bt = tb[t * 16 + l15];
        v2f bfrag;
        // lanes 0-15: K=0 row -> 1, K=1 row -> -2b ; lanes 16-31: K=2 -> b^2, K=3 -> 0
        bfrag.x = lo ? 1.0f        : bt * bt;
        bfrag.y = lo ? -2.0f * bt  : 0.0f;

        unsigned int colMin = FINF_BITS;
        #pragma unroll
        for (int k = 0; k < NTILE_C; ++k) {
            v8f d = __builtin_amdgcn_wmma_f32_16x16x4_f32(
                false, afrag[k], false, bfrag, (short)0, czero, false, false);
            #pragma unroll
            for (int r = 0; r < 8; ++r) {
                unsigned int u = __float_as_uint(d[r]);   // d >= 0: uint order == float order
                colMin = (u < colMin) ? u : colMin;                    // v_min_u32
                unsigned int ra = rowAcc[k * 8 + r];
                rowAcc[k * 8 + r] = (u < ra) ? u : ra;                 // v_min_u32
            }
        }
        // lane j has min over rows 0-7 (lo) / 8-15 (hi): merge the halves
        unsigned int other = (unsigned int)__shfl_xor((int)colMin, 16, 32);
        colMin = (other < colMin) ? other : colMin;
        sumMin += __uint_as_float(colMin);   // each target counted in lane j and j+16
    }

    // wave-sum target->center mins (each target counted twice -> *0.5)
    #pragma unroll
    for (int m = 16; m >= 1; m >>= 1)
        sumMin += __shfl_xor(sumMin, m, 32);
    if (lane == 0)
        atomicAdd((float*)&ws[4 + b], 0.5f * sumMin);

    // row (center) minima: butterfly within each 16-lane half, atomic-min to ws
    unsigned int* cmin = ws + WS_CMIN + b * NB;
    #pragma unroll
    for (int k = 0; k < NTILE_C; ++k) {
        #pragma unroll
        for (int r = 0; r < 8; ++r) {
            unsigned int v = rowAcc[k * 8 + r];
            #pragma unroll
            for (int m = 8; m >= 1; m >>= 1) {
                unsigned int o = (unsigned int)__shfl_xor((int)v, m, 32); // stays in-half
                v = (o < v) ? o : v;
            }
            if (l15 == 0) {
                int row = k * 16 + (lo ? r : 8 + r);
                atomicMin(&cmin[row], v);
            }
        }
    }
}

// ---------------------------------------------------------------------------
// Masked reductions for RMSE + SILog.
// ---------------------------------------------------------------------------
__global__ __launch_bounds__(256)
void stats_kernel(const float* __restrict__ pred,
                  const float* __restrict__ targ,
                  const unsigned char* __restrict__ mask,
                  unsigned int* __restrict__ ws, int n) {
    float cnt = 0.f, sq = 0.f, sd = 0.f, sd2 = 0.f;
    for (int i = blockIdx.x * blockDim.x + threadIdx.x; i < n;
         i += blockDim.x * gridDim.x) {
        float m = mask[i] ? 1.0f : 0.0f;
        float p = pred[i], t = targ[i];
        float diff = p - t;
        float d = logf(p + EPS_F) - logf(t + EPS_F);
        cnt += m;
        sq  += m * diff * diff;
        sd  += m * d;
        sd2 += m * d * d;
    }
    #pragma unroll
    for (int m = 16; m >= 1; m >>= 1) {
        cnt += __shfl_xor(cnt, m, 32);
        sq  += __shfl_xor(sq , m, 32);
        sd  += __shfl_xor(sd , m, 32);
        sd2 += __shfl_xor(sd2, m, 32);
    }
    if ((threadIdx.x & 31) == 0) {
        atomicAdd((float*)&ws[0], cnt);
        atomicAdd((float*)&ws[1], sq);
        atomicAdd((float*)&ws[2], sd);
        atomicAdd((float*)&ws[3], sd2);
    }
}

// ---------------------------------------------------------------------------
// Finalize: chamfer mean + RMSE + SILog -> d_out[0].
// ---------------------------------------------------------------------------
__global__ __launch_bounds__(256)
void finalize_kernel(const unsigned int* __restrict__ ws, float* __restrict__ out) {
    __shared__ float sh[256];
    float local = 0.0f;
    for (int j = threadIdx.x; j < BATCH * NB; j += blockDim.x)
        local += __uint_as_float(ws[WS_CMIN + j]);   // sum of per-center mins
    sh[threadIdx.x] = local;
    __syncthreads();
    for (int s = 128; s > 0; s >>= 1) {
        if (threadIdx.x < s) sh[threadIdx.x] += sh[threadIdx.x + s];
        __syncthreads();
    }
    if (threadIdx.x == 0) {
        float cham = sh[0];
        for (int bb = 0; bb < BATCH; ++bb)
            cham += __uint_as_float(ws[4 + bb]);     // target->center partials
        cham /= (float)BATCH;                        // mean over batch

        float cnt = __uint_as_float(ws[0]);
        float sq  = __uint_as_float(ws[1]);
        float sd  = __uint_as_float(ws[2]);
        float sd2 = __uint_as_float(ws[3]);
        float l2    = sqrtf(sq / cnt);
        float dmean = sd / cnt;
        float silog = 10.0f * sqrtf(sd2 / cnt - 0.85f * dmean * dmean);
        out[0] = l2 + silog + cham;
    }
}

extern "C" void kernel_launch(void* const* d_in, const int* in_sizes, int n_in,
                              void* d_out, int out_size, void* d_ws, size_t ws_size,
                              hipStream_t stream) {
    const float*         pred  = (const float*)d_in[0];
    const float*         targ  = (const float*)d_in[1];
    const float*         edges = (const float*)d_in[2];
    const unsigned char* mask  = (const unsigned char*)d_in[3];
    unsigned int* ws  = (unsigned int*)d_ws;
    float*        out = (float*)d_out;
    const int n = in_sizes[0];   // B*H*W = 614400

    init_ws_kernel<<<1, 256, 0, stream>>>(ws);
    dim3 cgrid(BATCH, 160);                       // 160 waves/batch, 30 tiles each
    chamfer_kernel<<<cgrid, 32, 0, stream>>>(targ, edges, ws);
    stats_kernel<<<480, 256, 0, stream>>>(pred, targ, mask, ws, n);
    finalize_kernel<<<1, 256, 0, stream>>>(ws, out);
}